// RuleLearner_40690520162774
// MI455X (gfx1250) — compile-verified
//
#include <hip/hip_runtime.h>
#include <cmath>

typedef float v2f __attribute__((ext_vector_type(2)));
typedef float v8f __attribute__((ext_vector_type(8)));

#define SEQ   5
#define NSYM  9      // 1 + NUM_SYMS
#define NU    14     // SEQ + NSYM
#define NI    128    // invented rules
#define NB    4096   // batch
#define L2E   1.44269504088896f

// workspace layout (floats):
//  [0, 2048)                : cmap matrix, NI x 16 (cols >= 9 zero)
//  [2048, 2048 + NB*NI)     : sel matrix (softmax-normalized), NB x NI row-major
#define WS_CMAP_OFF 0
#define WS_SEL_OFF  (NI * 16)

__device__ __forceinline__ float sigm(float x) {
    return 1.0f / (1.0f + exp2f(-L2E * x));
}

__global__ __launch_bounds__(NI) void rl_main(
    const int*   __restrict__ inputs,
    const int*   __restrict__ inv_inputs,
    const int*   __restrict__ inv_labels,
    const float* __restrict__ inv_unaryp,
    const float* __restrict__ inv_binaryp,
    const float* __restrict__ inv_out_map,
    float*       __restrict__ out,
    float*       __restrict__ ws,
    float sm_scale)
{
    const int b = blockIdx.x;
    const int i = threadIdx.x;
    const float l2scale = sm_scale * L2E;

    // ---- per-invented-rule tables (recomputed per block; all cached in L2) ----
    int si[SEQ];
#pragma unroll
    for (int m = 0; m < SEQ; ++m) si[m] = inv_inputs[i * SEQ + m];

    float onePos[SEQ], oneSym[SEQ];
#pragma unroll
    for (int m = 0; m < SEQ; ++m) {
        onePos[m] = 1.0f - sigm(inv_unaryp[(i * SEQ + m) * NU + m]);
        oneSym[m] = 1.0f - sigm(inv_unaryp[(i * SEQ + m) * NU + SEQ + si[m]]);
    }

    float c[SEQ][SEQ];
#pragma unroll
    for (int l = 0; l < SEQ; ++l)
#pragma unroll
        for (int m = 0; m < SEQ; ++m) {
            float pv = sigm(inv_binaryp[(i * SEQ + l) * SEQ + m]);
            c[l][m] = (l != m && si[l] == si[m]) ? pv : 0.0f;
        }

    // out_map softmax over SEQ+1
    float om[SEQ + 1];
    {
        float mx = -1e30f;
#pragma unroll
        for (int j = 0; j < SEQ + 1; ++j) {
            om[j] = inv_out_map[i * (SEQ + 1) + j];
            mx = fmaxf(mx, om[j]);
        }
        float s = 0.0f;
#pragma unroll
        for (int j = 0; j < SEQ + 1; ++j) { om[j] = exp2f((om[j] - mx) * L2E); s += om[j]; }
        float inv = 1.0f / s;
#pragma unroll
        for (int j = 0; j < SEQ + 1; ++j) om[j] *= inv;
    }

    // block 0 publishes the GEMM B-operand: cmap[i][u] = om[SEQ] * onehot(label)
    if (b == 0) {
        int lab = inv_labels[i];
#pragma unroll
        for (int u = 0; u < 16; ++u)
            ws[WS_CMAP_OFF + i * 16 + u] = (u == lab) ? om[SEQ] : 0.0f;
    }

    // ---- batch-row features ----
    int sb[SEQ];
#pragma unroll
    for (int k = 0; k < SEQ; ++k) sb[k] = inputs[b * SEQ + k];

    // uni[m][k]: one-hot features collapse the U=14 product to two factors
    float uni[SEQ][SEQ];
#pragma unroll
    for (int m = 0; m < SEQ; ++m)
#pragma unroll
        for (int k = 0; k < SEQ; ++k) {
            float t = (k == m) ? 1.0f : onePos[m];
            t *= (sb[k] == si[m]) ? 1.0f : oneSym[m];
            uni[m][k] = t;
        }

    // s[m][k] = 1 - prod_{j!=k, sb[j]==sb[k]} (1 - uni[m][j])
    float sv[SEQ][SEQ];
#pragma unroll
    for (int m = 0; m < SEQ; ++m)
#pragma unroll
        for (int k = 0; k < SEQ; ++k) {
            float p = 1.0f;
#pragma unroll
            for (int j = 0; j < SEQ; ++j)
                if (j != k)
                    p *= (sb[j] == sb[k]) ? (1.0f - uni[m][j]) : 1.0f;
            sv[m][k] = 1.0f - p;
        }

    // uni_sets[l][k] = uni[l][k] * prod_m (1 - c[l][m]*(1 - s[m][k]))
    float us[SEQ][SEQ];
#pragma unroll
    for (int l = 0; l < SEQ; ++l)
#pragma unroll
        for (int k = 0; k < SEQ; ++k) {
            float p = 1.0f;
#pragma unroll
            for (int m = 0; m < SEQ; ++m)
                p *= 1.0f - c[l][m] * (1.0f - sv[m][k]);
            us[l][k] = uni[l][k] * p;
        }

    // node_select softmax over k (args in [0, sm_scale] -> no max shift needed),
    // folded with out_map: w[k] = sum_l om[l] * softmax_k(us[l]*scale)[k]
    float w[SEQ] = {0.0f, 0.0f, 0.0f, 0.0f, 0.0f};
#pragma unroll
    for (int l = 0; l < SEQ; ++l) {
        float e[SEQ]; float se = 0.0f;
#pragma unroll
        for (int k = 0; k < SEQ; ++k) { e[k] = exp2f(us[l][k] * l2scale); se += e[k]; }
        float f = om[l] / se;
#pragma unroll
        for (int k = 0; k < SEQ; ++k) w[k] += e[k] * f;
    }

    // inv_uni = prod_m (1 - prod_k (1 - us[m][k]))
    float iu = 1.0f;
#pragma unroll
    for (int m = 0; m < SEQ; ++m) {
        float p = 1.0f;
#pragma unroll
        for (int k = 0; k < SEQ; ++k) p *= 1.0f - us[m][k];
        iu *= 1.0f - p;
    }
    float E = exp2f(iu * l2scale);   // un-normalized inv_select numerator

    // ---- deterministic block reduction: S = sum_i E, g[k] = sum_i E*w[k] ----
    __shared__ float red[6 * NI];
    red[0 * NI + i] = E;
#pragma unroll
    for (int k = 0; k < SEQ; ++k) red[(k + 1) * NI + i] = E * w[k];
    __syncthreads();
    for (int off = NI / 2; off > 0; off >>= 1) {
        if (i < off) {
#pragma unroll
            for (int j = 0; j < 6; ++j) red[j * NI + i] += red[j * NI + i + off];
        }
        __syncthreads();
    }

    float inv = 1.0f / red[0];
    // normalized sel row for the WMMA GEMM kernel (coalesced store)
    ws[WS_SEL_OFF + b * NI + i] = E * inv;

    if (i == 0) {
        // scatter part of predictions: sum_k [sb[k]==u] * g[k] / S
#pragma unroll
        for (int u = 0; u < NSYM; ++u) {
            float p = 0.0f;
#pragma unroll
            for (int k = 0; k < SEQ; ++k)
                p += (sb[k] == u) ? red[(k + 1) * NI] : 0.0f;
            out[b * NSYM + u] = p * inv;
        }
    }
}

// predictions += sel(4096x128) @ cmap(128x16), one wave per 16x16 tile,
// K=128 chained through 32x V_WMMA_F32_16X16X4_F32.
__global__ __launch_bounds__(32) void rl_gemm(const float* __restrict__ ws,
                                              float* __restrict__ out)
{
    const float* sel  = ws + WS_SEL_OFF;
    const float* cmap = ws + WS_CMAP_OFF;

    const int lane = threadIdx.x;
    const int half = lane >> 4;        // 0: K pair {0,1}; 1: K pair {2,3}
    const int l15  = lane & 15;
    const int b0   = blockIdx.x * 16;

    const int arow = b0 + l15;         // A matrix 16x4: lanes 0-15 / 16-31 both hold M=0..15
    const int koff = half * 2;

    v8f acc = {};
#pragma unroll
    for (int kk = 0; kk < NI; kk += 4) {
        v2f a;                         // A VGPR0/1 = K=koff, koff+1 for this half-wave
        a.x = sel[arow * NI + kk + koff + 0];
        a.y = sel[arow * NI + kk + koff + 1];
        v2f bv;                        // B 4x16: row K=koff(+1), N = l15
        bv.x = cmap[(kk + koff + 0) * 16 + l15];
        bv.y = cmap[(kk + koff + 1) * 16 + l15];
        acc = __builtin_amdgcn_wmma_f32_16x16x4_f32(
            /*neg_a=*/false, a, /*neg_b=*/false, bv,
            /*c_mod=*/(short)0, acc, /*reuse_a=*/false, /*reuse_b=*/false);
    }

    // C/D layout: VGPR r -> M = r + 8*half, N = l15
    const int n = l15;
    if (n < NSYM) {
#pragma unroll
        for (int r = 0; r < 8; ++r) {
            int bb = b0 + half * 8 + r;
            out[bb * NSYM + n] += acc[r];
        }
    }
}

extern "C" void kernel_launch(void* const* d_in, const int* in_sizes, int n_in,
                              void* d_out, int out_size, void* d_ws, size_t ws_size,
                              hipStream_t stream) {
    const int*   inputs      = (const int*)  d_in[0];
    const int*   inv_inputs  = (const int*)  d_in[1];
    const int*   inv_labels  = (const int*)  d_in[2];
    const float* inv_unaryp  = (const float*)d_in[3];
    const float* inv_binaryp = (const float*)d_in[4];
    const float* inv_out_map = (const float*)d_in[5];
    float* out = (float*)d_out;
    float* ws  = (float*)d_ws;   // needs (NI*16 + NB*NI)*4 ~= 2.1 MB

    const float sm_scale = logf(0.999f * (float)(NI - 1) / (1.0f - 0.999f));

    rl_main<<<NB, NI, 0, stream>>>(inputs, inv_inputs, inv_labels,
                                   inv_unaryp, inv_binaryp, inv_out_map,
                                   out, ws, sm_scale);
    rl_gemm<<<NB / 16, 32, 0, stream>>>(ws, out);
}